// MeanAggregatorSparse_54863912239180
// MI455X (gfx1250) — compile-verified
//
#include <hip/hip_runtime.h>

#define N_NODES 50000
#define N_EDGES 640000
#define D_FEAT  128
#define D_IN    256            // 2 * D_FEAT (concat)
#define D_OUT   128
#define ROW_TILES (N_NODES / 16)   // 3125, exact
#define APITCH  260            // LDS pitch in floats: bank = (4*row+k)%64 -> conflict-free

typedef __attribute__((ext_vector_type(2))) float v2f;
typedef __attribute__((ext_vector_type(4))) float v4f;
typedef __attribute__((ext_vector_type(8))) float v8f;

// ---------------------------------------------------------------------------
// Kernel 1: zero seg_sum + counts. Regular (RT) stores on purpose: primes L2
// with the seg_sum lines so the atomic storm in kernel 2 hits L2 immediately.
// ---------------------------------------------------------------------------
__global__ __launch_bounds__(256) void zero_ws_kernel(float* __restrict__ ws, int n4) {
    int i = blockIdx.x * 256 + threadIdx.x;
    if (i < n4) {
        v4f z = {0.f, 0.f, 0.f, 0.f};
        ((v4f*)ws)[i] = z;
    }
}

// ---------------------------------------------------------------------------
// Kernel 2: scatter-add. One wave (32 lanes) per edge; v4f per lane.
//  - nbr_feat (327MB, read once) loaded non-temporally so it does not evict
//    the L2-resident seg_sum (25.6MB << 192MB L2) that the atomics hammer.
//  - unsafeAtomicAdd guarantees single-shot global_atomic_add_f32 (no CAS
//    loop); result unused -> non-returning variant; wave's 128 atomics span
//    one contiguous 512B row -> coalesces to 8 line-granular L2 transactions.
// ---------------------------------------------------------------------------
__global__ __launch_bounds__(256) void scatter_add_kernel(const float* __restrict__ nbr,
                                                          const int*   __restrict__ idx,
                                                          float* __restrict__ seg,
                                                          float* __restrict__ cnt) {
    long long t = (long long)blockIdx.x * 256 + threadIdx.x;
    int e    = (int)(t >> 5);
    int lane = (int)(t & 31);
    if (e >= N_EDGES) return;

    int n = idx[e];
    const v4f v = __builtin_nontemporal_load(
        (const v4f*)(nbr + (long long)e * D_FEAT) + lane);
    float* dst = seg + (long long)n * D_FEAT + lane * 4;
    unsafeAtomicAdd(dst + 0, v.x);
    unsafeAtomicAdd(dst + 1, v.y);
    unsafeAtomicAdd(dst + 2, v.z);
    unsafeAtomicAdd(dst + 3, v.w);
    if (lane == 0) unsafeAtomicAdd(cnt + n, 1.0f);
}

// ---------------------------------------------------------------------------
// Kernel 3: out[n][o] = sum_k h[n][k] * W[o][k],  h = [self || seg/max(cnt,1)]
// 8 waves/WG; wave w owns output cols [16w,16w+16). B fragments (W) in regs.
// A tile (16 rows x 256 K) staged in LDS per row-tile; V_WMMA_F32_16X16X4_F32.
// Fragment layout (ISA 7.12.2): lane = m + 16g;
//   A vgpr v  = A[m][k + 2g + v]
//   B vgpr v  = B[k + 2g + v][m] = W[colBase+m][k + 2g + v]
//   D vgpr v  -> row (v + 8g), col m
// ---------------------------------------------------------------------------
__global__ __launch_bounds__(256) void gemm_wmma_kernel(const float* __restrict__ selfF,
                                                        const float* __restrict__ seg,
                                                        const float* __restrict__ cnt,
                                                        const float* __restrict__ W,
                                                        float* __restrict__ out) {
    __shared__ float sA[16][APITCH];   // 16.6 KB

    const int tid     = threadIdx.x;
    const int wave    = tid >> 5;
    const int lane    = tid & 31;
    const int m       = lane & 15;
    const int g       = lane >> 4;
    const int colBase = wave * 16;

    // Preload this wave's 16 rows of W (the B matrix) into registers:
    // 64 k-steps x 2 floats/lane = 128 VGPRs. Loaded once per WG, reused
    // across all row tiles this WG processes (W stays L2-resident: 128KB).
    v2f bfrag[64];
    {
        const float* wrow = W + (long long)(colBase + m) * D_IN;
#pragma unroll
        for (int ks = 0; ks < 64; ++ks) {
            int kk = 4 * ks + 2 * g;
            bfrag[ks] = *(const v2f*)(wrow + kk);
        }
    }

    for (int rt = blockIdx.x; rt < ROW_TILES; rt += gridDim.x) {
        const int rowBase = rt * 16;

        // Cooperative A-tile load: 16 rows x 256 floats = 1024 v4f chunks.
        // Cols [0,128): self_feat (streamed once -> NT load);
        // cols [128,256): seg_sum * 1/max(count,1) (L2-hot after scatter).
        for (int i = tid; i < 1024; i += 256) {
            int row = i >> 6;            // 64 v4f per row
            int c4  = (i & 63) << 2;     // starting float column
            int gr  = rowBase + row;
            v4f val;
            if (c4 < D_FEAT) {
                val = __builtin_nontemporal_load(
                    (const v4f*)(selfF + (long long)gr * D_FEAT + c4));
            } else {
                float c   = cnt[gr];
                float inv = 1.0f / fmaxf(c, 1.0f);
                v4f   s   = *(const v4f*)(seg + (long long)gr * D_FEAT + (c4 - D_FEAT));
                val = s * inv;
            }
            *(v4f*)&sA[row][c4] = val;   // pitch 1040B: 16B-aligned stores
        }
        __syncthreads();

        v8f acc = {};   // fresh accumulator per tile
#pragma unroll
        for (int ks = 0; ks < 64; ++ks) {
            int kk = 4 * ks + 2 * g;
            v2f a = *(const v2f*)&sA[m][kk];   // conflict-free: bank = (4m+kk)%64
            acc = __builtin_amdgcn_wmma_f32_16x16x4_f32(
                      /*neg_a=*/false, a,
                      /*neg_b=*/false, bfrag[ks],
                      /*c_mod=*/(short)0, acc,
                      /*reuse_a=*/false, /*reuse_b=*/false);
        }

        // D vgpr v -> out[rowBase + v + 8g][colBase + m]; written once -> NT.
        float* op = out + (long long)(rowBase + 8 * g) * D_OUT + colBase + m;
#pragma unroll
        for (int v = 0; v < 8; ++v)
            __builtin_nontemporal_store(acc[v], op + (long long)v * D_OUT);

        __syncthreads();   // protect sA before next tile overwrites it
    }
}

// ---------------------------------------------------------------------------
extern "C" void kernel_launch(void* const* d_in, const int* in_sizes, int n_in,
                              void* d_out, int out_size, void* d_ws, size_t ws_size,
                              hipStream_t stream) {
    const float* selfF = (const float*)d_in[0];   // [50000,128] f32
    const float* nbr   = (const float*)d_in[1];   // [640000,128] f32
    const int*   idx   = (const int*)  d_in[2];   // [640000] int
    const float* W     = (const float*)d_in[3];   // [128,256] f32
    float* out = (float*)d_out;                   // [50000,128] f32

    float* seg = (float*)d_ws;                          // 50000*128 f32
    float* cnt = seg + (size_t)N_NODES * D_FEAT;        // 50000 f32

    // 1) zero workspace: (6,400,000 + 50,000) floats = 1,612,500 v4f
    const int n4 = (N_NODES * D_FEAT + N_NODES) / 4;
    zero_ws_kernel<<<(n4 + 255) / 256, 256, 0, stream>>>((float*)d_ws, n4);

    // 2) scatter-add: one wave per edge -> 640000*32 threads / 256 = 80000 blocks
    scatter_add_kernel<<<(N_EDGES * 32) / 256, 256, 0, stream>>>(nbr, idx, seg, cnt);

    // 3) WMMA GEMM: 1024 WGs grid-stride over 3125 row tiles (amortizes W preload)
    gemm_wmma_kernel<<<1024, 256, 0, stream>>>(selfF, seg, cnt, W, out);
}